// TripletLoss_65481071411076
// MI455X (gfx1250) — compile-verified
//
#include <hip/hip_runtime.h>

typedef __attribute__((ext_vector_type(2))) float v2f;
typedef __attribute__((ext_vector_type(8))) float v8f;

#define N 512
#define D 128
#define FLT_BIG 3.402823466e+38f

// ---------------------------------------------------------------------------
// Kernel 1: per-row sum of squares  sq[r] = sum_k E[r][k]^2
// ---------------------------------------------------------------------------
__global__ void sq_kernel(const float* __restrict__ E, float* __restrict__ sq) {
    int row = blockIdx.x * blockDim.x + threadIdx.x;
    if (row < N) {
        const float* e = E + row * D;
        float s = 0.0f;
#pragma unroll 8
        for (int k = 0; k < D; ++k) { float v = e[k]; s = fmaf(v, v, s); }
        sq[row] = s;
    }
}

// ---------------------------------------------------------------------------
// Kernel 2: Gram via V_WMMA_F32_16X16X4_F32, then pdist = sq_i + sq_j - 2G,
// clamped at 0, diagonal zeroed.  One wave (32 threads) per 16x16 tile.
//
// fp32 A (16x4) lane layout: lanes 0-15 -> row m=lane, v0=k0, v1=k0+1;
//                            lanes 16-31 -> row m=lane-16, v0=k0+2, v1=k0+3.
// B (4x16) mirrors it with n = lane&15, so for G = E*E^T both tiles load the
// same pattern from different row blocks of E.
// C/D layout: element (M = v + 8*(lane>>4), N = lane&15) in VGPR v.
// ---------------------------------------------------------------------------
__global__ void gram_pdist_kernel(const float* __restrict__ E,
                                  const float* __restrict__ sq,
                                  float* __restrict__ pdist) {
    const int tile = blockIdx.x;      // 0..1023
    const int ti = tile >> 5;         // row-tile index (0..31)
    const int tj = tile & 31;         // col-tile index (0..31)
    const int lane = threadIdx.x;     // 0..31 (wave32, EXEC all ones)
    const int half = lane >> 4;       // 0 or 1
    const int r = lane & 15;

    const float* arow = E + (ti * 16 + r) * D + 2 * half;
    const float* brow = E + (tj * 16 + r) * D + 2 * half;

    v8f c = {};
#pragma unroll 4
    for (int k0 = 0; k0 < D; k0 += 4) {
        v2f a, b;
        a.x = arow[k0];
        a.y = arow[k0 + 1];
        b.x = brow[k0];
        b.y = brow[k0 + 1];
        // 8 args: (neg_a, A, neg_b, B, c_mod, C, reuse_a, reuse_b)
        c = __builtin_amdgcn_wmma_f32_16x16x4_f32(
            false, a, false, b, (short)0, c, false, false);
    }

    const int col = tj * 16 + r;
    const float sqc = sq[col];
#pragma unroll
    for (int v = 0; v < 8; ++v) {
        const int row = ti * 16 + v + 8 * half;
        float d = sq[row] + sqc - 2.0f * c[v];
        d = fmaxf(d, 0.0f);
        if (row == col) d = 0.0f;
        pdist[row * N + col] = d;
    }
}

// ---------------------------------------------------------------------------
// Kernel 3: per-anchor row stats. One wave per anchor row.
//   row_min  = min_j pdist[a][j]                     (diag 0 -> effectively 0)
//   nin      = max over j with label[j]!=label[a] of pdist[a][j]  (-inf if none)
//   neg_inside[a] = max(nin, row_min)   ==  max_j((d - rmin)*adjnot) + rmin
// ---------------------------------------------------------------------------
__global__ void row_stats_kernel(const float* __restrict__ pdist,
                                 const int* __restrict__ label,
                                 float* __restrict__ neg_inside) {
    const int a = blockIdx.x;
    const int lane = threadIdx.x;      // 0..31
    const int la = label[a];
    const float* row = pdist + a * N;

    float mn  = FLT_BIG;
    float nin = -FLT_BIG;
    for (int j = lane; j < N; j += 32) {
        const float d = row[j];
        mn = fminf(mn, d);
        if (label[j] != la) nin = fmaxf(nin, d);
    }
#pragma unroll
    for (int off = 16; off > 0; off >>= 1) {
        mn  = fminf(mn,  __shfl_xor(mn,  off, 32));
        nin = fmaxf(nin, __shfl_xor(nin, off, 32));
    }
    if (lane == 0) neg_inside[a] = fmaxf(nin, mn);
}

// ---------------------------------------------------------------------------
// Kernel 4: loss accumulation. One 256-thread block per anchor a.
// Only positive pairs (same label, p!=a) contribute; for each, scan the LDS-
// resident pdist row for the semi-hard negative:
//   min over k of pdist[a][k] s.t. label[k]!=label[a] and pdist[a][k] > t.
// Deterministic per-block tree reduce; partials go to fixed slots.
// ---------------------------------------------------------------------------
__global__ void loss_kernel(const float* __restrict__ pdist,
                            const int* __restrict__ label,
                            const float* __restrict__ neg_inside,
                            float* __restrict__ psum,
                            float* __restrict__ pcnt) {
    __shared__ float row[N];
    __shared__ int   lab[N];
    __shared__ float rs[256];
    __shared__ float rc[256];

    const int a = blockIdx.x;
    const int tid = threadIdx.x;

    for (int j = tid; j < N; j += 256) {
        row[j] = pdist[a * N + j];
        lab[j] = label[j];
    }
    __syncthreads();

    const int la = lab[a];
    const float nin = neg_inside[a];

    float lsum = 0.0f;
    float lcnt = 0.0f;
    for (int p = tid; p < N; p += 256) {
        if (p == a || lab[p] != la) continue;   // mask_pos == 0: exact skip
        lcnt += 1.0f;
        const float t = row[p];
        float m = FLT_BIG;
        int found = 0;
        for (int k = 0; k < N; ++k) {
            const float d = row[k];
            if (lab[k] != la && d > t) { m = fminf(m, d); found = 1; }
        }
        const float semi = found ? m : nin;
        lsum += fmaxf(1.0f + t - semi, 0.0f);   // MARGIN = 1.0
    }

    rs[tid] = lsum;
    rc[tid] = lcnt;
    __syncthreads();
#pragma unroll
    for (int off = 128; off > 0; off >>= 1) {
        if (tid < off) { rs[tid] += rs[tid + off]; rc[tid] += rc[tid + off]; }
        __syncthreads();
    }
    if (tid == 0) { psum[a] = rs[0]; pcnt[a] = rc[0]; }
}

// ---------------------------------------------------------------------------
// Kernel 5: fixed-order final reduction -> scalar loss (deterministic).
// ---------------------------------------------------------------------------
__global__ void finalize_kernel(const float* __restrict__ psum,
                                const float* __restrict__ pcnt,
                                float* __restrict__ out) {
    if (blockIdx.x == 0 && threadIdx.x == 0) {
        float s = 0.0f, c = 0.0f;
        for (int i = 0; i < N; ++i) { s += psum[i]; c += pcnt[i]; }
        out[0] = s / c;
    }
}

// ---------------------------------------------------------------------------
extern "C" void kernel_launch(void* const* d_in, const int* in_sizes, int n_in,
                              void* d_out, int out_size, void* d_ws, size_t ws_size,
                              hipStream_t stream) {
    const float* E     = (const float*)d_in[0];   // [512,128] fp32
    const int*   label = (const int*)d_in[1];     // [512] int32
    float*       out   = (float*)d_out;           // scalar fp32

    float* ws         = (float*)d_ws;
    float* pdist      = ws;                 // N*N floats (1 MB)
    float* sq         = pdist + N * N;      // N
    float* neg_inside = sq + N;             // N
    float* psum       = neg_inside + N;     // N
    float* pcnt       = psum + N;           // N

    sq_kernel<<<(N + 255) / 256, 256, 0, stream>>>(E, sq);
    gram_pdist_kernel<<<(N / 16) * (N / 16), 32, 0, stream>>>(E, sq, pdist);
    row_stats_kernel<<<N, 32, 0, stream>>>(pdist, label, neg_inside);
    loss_kernel<<<N, 256, 0, stream>>>(pdist, label, neg_inside, psum, pcnt);
    finalize_kernel<<<1, 1, 0, stream>>>(psum, pcnt, out);
}